// HybridGaussianLayer_18786186953013
// MI455X (gfx1250) — compile-verified
//
#include <hip/hip_runtime.h>
#include <hip/hip_bf16.h>

#define B_DIM   64
#define IN_DIM  128
#define OUT_DIM 256
#define EPS     1e-8f
#define LOG2E   1.4426950408889634f

typedef __attribute__((ext_vector_type(2))) float v2f;
typedef __attribute__((ext_vector_type(8))) float v8f;

// ---------------------------------------------------------------------------
// Kernel 1: linear_out = x @ W.T + bias  via V_WMMA_F32_16X16X4_F32 (CDNA5).
// One wave32 per 16x16 output tile. Grid = (64/16) * (256/16) = 64 waves.
//
// ISA VGPR layouts (CDNA5 ISA):
//   A 16x4 f32 : lanes 0-15 -> M=lane, VGPR0=K0, VGPR1=K1
//                lanes 16-31 -> M=lane-16, VGPR0=K2, VGPR1=K3
//   B 4x16 f32 : lanes 0-15 -> N=lane, VGPR r = K r;  lanes 16-31 -> K r+2
//   C/D 16x16  : VGPR r: lanes 0-15 M=r, lanes 16-31 M=r+8, N=lane%16
// ---------------------------------------------------------------------------
__global__ __launch_bounds__(32) void hgl_linear_wmma(
    const float* __restrict__ x,     // [B, IN]
    const float* __restrict__ W,     // [OUT, IN]
    const float* __restrict__ bias,  // [OUT]
    float* __restrict__ lin)         // [B, OUT]
{
    const int tile = blockIdx.x;
    const int tm   = tile % (B_DIM / 16);   // batch-row tile
    const int tn   = tile / (B_DIM / 16);   // out-col tile
    const int lane = threadIdx.x;           // 0..31 (wave32)
    const int half = lane >> 4;             // 0: K{0,1} ; 1: K{2,3}
    const int l    = lane & 15;

    const int row = tm * 16 + l;            // M index this lane provides for A
    const int col = tn * 16 + l;            // N index this lane provides for B

    const float* xrow = x + (size_t)row * IN_DIM;
    const float* wrow = W + (size_t)col * IN_DIM;   // B[k][n] = W[n][k]

    v8f acc = {};
    #pragma unroll
    for (int k = 0; k < IN_DIM; k += 4) {
        const int ka = k + half * 2;
        v2f a = *(const v2f*)(xrow + ka);   // K=ka, ka+1 for row M
        v2f b = *(const v2f*)(wrow + ka);   // K=ka, ka+1 for col N
        acc = __builtin_amdgcn_wmma_f32_16x16x4_f32(
            /*neg_a=*/false, a, /*neg_b=*/false, b,
            /*c_mod=*/(short)0, acc, /*reuse_a=*/false, /*reuse_b=*/false);
    }

    const float bv = bias[col];
    #pragma unroll
    for (int r = 0; r < 8; ++r) {
        const int m = tm * 16 + r + half * 8;
        lin[(size_t)m * OUT_DIM + col] = acc[r] + bv;
    }
}

// ---------------------------------------------------------------------------
// Kernel 2: Gaussian path. One wave32 per (b, o) pair; 16384 waves total.
//
// Hot-loop algebra:  cl*(zi-zj)^2 = a_i + b_j + m_i*z_j  with
//   cl  = -log2(e) / (2*exp(log_sigma)^2 + EPS)   (exp2-folded scale)
//   a_i = cl*z_i^2 (register), m_i = -2*cl*z_i (register), b_j = cl*z_j^2 (LDS)
// Per element: v_add + v_fma + v_exp_f32 (native exp2) + v_add
//   -> 3 VALU + 1 TRANS, co-executing; exp pipe is the hardware floor.
// ---------------------------------------------------------------------------
__global__ __launch_bounds__(32) void hgl_gaussian(
    const float* __restrict__ x,         // [B, IN]
    const float* __restrict__ W,         // [OUT, IN]
    const float* __restrict__ log_sigma, // [OUT]
    const float* __restrict__ alpha,     // [OUT]
    const float* __restrict__ lin,       // [B, OUT]
    float* __restrict__ out)             // [B, OUT]
{
    const int bo   = blockIdx.x;
    const int o    = bo % OUT_DIM;
    const int b    = bo / OUT_DIM;
    const int lane = threadIdx.x;        // 0..31

    __shared__ float zsh[IN_DIM];        // z_j
    __shared__ float bsh[IN_DIM];        // cl * z_j^2

    // exp2-folded Gaussian scale (uniform across wave)
    const float es = __expf(log_sigma[o]);
    const float cl = -LOG2E / (2.0f * es * es + EPS);

    // z_i = x[b,i] * W[o,i] ; lane covers i = 4*lane .. 4*lane+3
    const float4 xv = reinterpret_cast<const float4*>(x + (size_t)b * IN_DIM)[lane];
    const float4 wv = reinterpret_cast<const float4*>(W + (size_t)o * IN_DIM)[lane];
    const float z0 = xv.x * wv.x;
    const float z1 = xv.y * wv.y;
    const float z2 = xv.z * wv.z;
    const float z3 = xv.w * wv.w;

    reinterpret_cast<float4*>(zsh)[lane] = make_float4(z0, z1, z2, z3);
    reinterpret_cast<float4*>(bsh)[lane] =
        make_float4(cl * z0 * z0, cl * z1 * z1, cl * z2 * z2, cl * z3 * z3);
    __syncthreads();

    // per-lane quadratic coefficients for the 4 owned i-values
    const float a0 = cl * z0 * z0, m0 = -2.0f * cl * z0;
    const float a1 = cl * z1 * z1, m1 = -2.0f * cl * z1;
    const float a2 = cl * z2 * z2, m2 = -2.0f * cl * z2;
    const float a3 = cl * z3 * z3, m3 = -2.0f * cl * z3;

    float s0 = 0.f, s1 = 0.f, s2 = 0.f, s3 = 0.f;
    #pragma unroll 4
    for (int j = 0; j < IN_DIM; j += 4) {
        const float4 zj = *reinterpret_cast<const float4*>(&zsh[j]);  // broadcast
        const float4 bj = *reinterpret_cast<const float4*>(&bsh[j]);  // broadcast
        s0 += __builtin_amdgcn_exp2f(__builtin_fmaf(m0, zj.x, a0 + bj.x));
        s0 += __builtin_amdgcn_exp2f(__builtin_fmaf(m0, zj.y, a0 + bj.y));
        s0 += __builtin_amdgcn_exp2f(__builtin_fmaf(m0, zj.z, a0 + bj.z));
        s0 += __builtin_amdgcn_exp2f(__builtin_fmaf(m0, zj.w, a0 + bj.w));
        s1 += __builtin_amdgcn_exp2f(__builtin_fmaf(m1, zj.x, a1 + bj.x));
        s1 += __builtin_amdgcn_exp2f(__builtin_fmaf(m1, zj.y, a1 + bj.y));
        s1 += __builtin_amdgcn_exp2f(__builtin_fmaf(m1, zj.z, a1 + bj.z));
        s1 += __builtin_amdgcn_exp2f(__builtin_fmaf(m1, zj.w, a1 + bj.w));
        s2 += __builtin_amdgcn_exp2f(__builtin_fmaf(m2, zj.x, a2 + bj.x));
        s2 += __builtin_amdgcn_exp2f(__builtin_fmaf(m2, zj.y, a2 + bj.y));
        s2 += __builtin_amdgcn_exp2f(__builtin_fmaf(m2, zj.z, a2 + bj.z));
        s2 += __builtin_amdgcn_exp2f(__builtin_fmaf(m2, zj.w, a2 + bj.w));
        s3 += __builtin_amdgcn_exp2f(__builtin_fmaf(m3, zj.x, a3 + bj.x));
        s3 += __builtin_amdgcn_exp2f(__builtin_fmaf(m3, zj.y, a3 + bj.y));
        s3 += __builtin_amdgcn_exp2f(__builtin_fmaf(m3, zj.z, a3 + bj.z));
        s3 += __builtin_amdgcn_exp2f(__builtin_fmaf(m3, zj.w, a3 + bj.w));
    }

    // L1 norm of support over i, and sum_i support_i * z_i (pre-normalization)
    float pn = fabsf(s0) + fabsf(s1) + fabsf(s2) + fabsf(s3);
    float pg = s0 * z0 + s1 * z1 + s2 * z2 + s3 * z3;

    #pragma unroll
    for (int off = 16; off > 0; off >>= 1) {       // wave32 reduction
        pn += __shfl_xor(pn, off, 32);
        pg += __shfl_xor(pg, off, 32);
    }

    if (lane == 0) {
        const float norm = fmaxf(pn, 1e-12f);
        const float g    = pg / norm;
        const float a    = 1.0f / (1.0f + __expf(-alpha[o]));
        out[(size_t)b * OUT_DIM + o] = a * g + (1.0f - a) * lin[(size_t)b * OUT_DIM + o];
    }
}

extern "C" void kernel_launch(void* const* d_in, const int* in_sizes, int n_in,
                              void* d_out, int out_size, void* d_ws, size_t ws_size,
                              hipStream_t stream) {
    const float* x         = (const float*)d_in[0];  // [64,128]
    const float* W         = (const float*)d_in[1];  // [256,128]
    const float* bias      = (const float*)d_in[2];  // [256]
    const float* log_sigma = (const float*)d_in[3];  // [256]
    const float* alpha     = (const float*)d_in[4];  // [256]
    float*       out       = (float*)d_out;          // [64,256]
    float*       lin       = (float*)d_ws;           // [64,256] scratch (64 KB)

    // Linear path via f32 WMMA: 64 tiles of 16x16.
    hgl_linear_wmma<<<(B_DIM / 16) * (OUT_DIM / 16), 32, 0, stream>>>(x, W, bias, lin);

    // Gaussian path + final blend: one wave per (b, o).
    hgl_gaussian<<<B_DIM * OUT_DIM, 32, 0, stream>>>(x, W, log_sigma, alpha, lin, out);
}